// PairBilinear_MixerBlock_83365315216075
// MI455X (gfx1250) — compile-verified
//
#include <hip/hip_runtime.h>
#include <hip/hip_bf16.h>
#include <stdint.h>

// ---------------------------------------------------------------------------
// Problem constants (from reference)
// ---------------------------------------------------------------------------
#define GG 8
#define DYC (2.0f / 7.0f)
#define DIM 1024
#define NUM_LAYERS 10
#define PP 512                         // DIM/2 pairs
#define TBL_FLOATS (PP * 2 * GG * GG)  // 65536 floats = 256 KB per layer
#define ROWS_PER_BLOCK 16              // y tile: 16*1024*4 = 64 KB LDS

typedef float v2f __attribute__((ext_vector_type(2)));
typedef float v8f __attribute__((ext_vector_type(8)));
typedef unsigned int u32x4 __attribute__((ext_vector_type(4)));
typedef int i32x8 __attribute__((ext_vector_type(8)));
typedef int i32x4 __attribute__((ext_vector_type(4)));

__device__ __forceinline__ float clip_dy(float d) {
    return fminf(fmaxf(d, -DYC), DYC);
}

// ---------------------------------------------------------------------------
// Kernel 1: Yeff = _project(Ys) via V_WMMA_F32_16X16X4_F32.
//
// Per 8x8 grid:  Yeff = 0.5 * (V0 @ S + L @ V1)
//   V0[r][0]=Y[r][0], V0[r][c]=clip(Y[r][c]-Y[r][c-1])   (row diffs)
//   V1[0][c]=Y[0][c], V1[r][c]=clip(Y[r][c]-Y[r-1][c])   (col diffs)
//   S upper-triangular ones (8x8), L lower-triangular ones (8x8)
// Four grids are packed 2x2 into one 16x16 tile; S,L become block-diagonal.
// K=16 contraction done as 4 chained K=4 WMMAs per product (8 WMMAs total).
// ---------------------------------------------------------------------------
__device__ __forceinline__ float Vval(const float* Gd, int r, int col) {
    int q = ((r >> 3) << 1) + (col >> 3);
    int rr = r & 7, c = col & 7;
    const float* g = Gd + q * 64 + rr * 8;
    return (c == 0) ? g[0] : clip_dy(g[c] - g[c - 1]);
}
__device__ __forceinline__ float V1val(const float* Gd, int j, int n) {
    int q = ((j >> 3) << 1) + (n >> 3);
    int r = j & 7, c = n & 7;
    const float* g = Gd + q * 64;
    return (r == 0) ? g[c] : clip_dy(g[r * 8 + c] - g[(r - 1) * 8 + c]);
}
__device__ __forceinline__ float Sval(int j, int n) {  // S_bd[j][n]
    return ((j >> 3) == (n >> 3) && (j & 7) <= (n & 7)) ? 1.0f : 0.0f;
}
__device__ __forceinline__ float Lval(int m, int j) {  // L_bd[m][j]
    return ((m >> 3) == (j >> 3) && (j & 7) <= (m & 7)) ? 1.0f : 0.0f;
}

__global__ __launch_bounds__(32) void project_wmma_kernel(
    const float* __restrict__ Ys, float* __restrict__ Yeff) {
    __shared__ float Gd[4 * 64];   // 4 grids of 8x8
    const int lane = threadIdx.x;  // 0..31, single wave => EXEC all ones
    const long tileBase = (long)blockIdx.x * 256;

#pragma unroll
    for (int k = 0; k < 8; ++k) Gd[lane * 8 + k] = Ys[tileBase + lane * 8 + k];
    __syncthreads();

    const int m = lane & 15;  // row (A) / col (B/D)
    const int h = lane >> 4;  // half-wave select

#if __has_builtin(__builtin_amdgcn_wmma_f32_16x16x4_f32)
    v8f acc = {};
#pragma unroll
    for (int k = 0; k < 4; ++k) {
        const int c0 = 4 * k + 2 * h;
        v2f a, b;
        // product 1: V0_tile @ S_bd   (A rows m, K-cols c0,c0+1; B K-rows c0,c0+1, col m)
        a.x = Vval(Gd, m, c0);
        a.y = Vval(Gd, m, c0 + 1);
        b.x = Sval(c0, m);
        b.y = Sval(c0 + 1, m);
        acc = __builtin_amdgcn_wmma_f32_16x16x4_f32(false, a, false, b,
                                                    (short)0, acc, false, false);
        // product 2: L_bd @ V1_tile
        a.x = Lval(m, c0);
        a.y = Lval(m, c0 + 1);
        b.x = V1val(Gd, c0, m);
        b.y = V1val(Gd, c0 + 1, m);
        acc = __builtin_amdgcn_wmma_f32_16x16x4_f32(false, a, false, b,
                                                    (short)0, acc, false, false);
    }
#pragma unroll
    for (int v = 0; v < 8; ++v) {
        const int M = v + 8 * h, N = m;
        const int q = ((M >> 3) << 1) + (N >> 3);
        Yeff[tileBase + q * 64 + (M & 7) * 8 + (N & 7)] = 0.5f * acc[v];
    }
#else
    // Scalar fallback (same math) — keeps file compiling if builtin absent.
#pragma unroll
    for (int v = 0; v < 8; ++v) {
        const int M = v + 8 * h, N = m;
        const int q = ((M >> 3) << 1) + (N >> 3);
        const int r = M & 7, c = N & 7;
        const float* g = Gd + q * 64;
        float f0 = g[r * 8 + 0];
        for (int cc = 1; cc <= c; ++cc) f0 += clip_dy(g[r * 8 + cc] - g[r * 8 + cc - 1]);
        float f1 = g[c];
        for (int rr = 1; rr <= r; ++rr) f1 += clip_dy(g[rr * 8 + c] - g[(rr - 1) * 8 + c]);
        Yeff[tileBase + q * 64 + r * 8 + c] = 0.5f * (f0 + f1);
    }
#endif
}

// ---------------------------------------------------------------------------
// Kernel 2: fused 10-layer mixer.
//   - y tile (16 rows x 1024) lives in LDS for the whole kernel.
//   - per layer, the 256 KB Yeff table is DMA'd into LDS with the Tensor
//     Data Mover (one 32 KB slice per wave), tracked by TENSORcnt.
//   - permutations are folded into pair indexing (they cancel), so each
//     layer is: load table -> barrier -> in-place bilinear residual update.
// LDS: 256 KB table + 64 KB y = 320 KB (full WGP LDS; one block per WGP).
// ---------------------------------------------------------------------------
__global__ __launch_bounds__(256) void mixer_kernel(
    const float* __restrict__ x, const float* __restrict__ bias,
    const float* __restrict__ Yeff, float* __restrict__ out) {
    extern __shared__ float smem[];
    float* tbl = smem;                // TBL_FLOATS
    float* y = smem + TBL_FLOATS;     // ROWS_PER_BLOCK * DIM

    const int tid = threadIdx.x;
    const long row0 = (long)blockIdx.x * ROWS_PER_BLOCK;

    // Phase 0: y = x + bias for this block's rows (float4 traffic).
    for (int v = tid; v < ROWS_PER_BLOCK * (DIM / 4); v += 256) {
        const int r = v >> 8, j4 = v & 255;
        const float4 xv = ((const float4*)(x + (row0 + r) * DIM))[j4];
        const float4 bv = ((const float4*)bias)[j4];
        float4 o;
        o.x = xv.x + bv.x; o.y = xv.y + bv.y;
        o.z = xv.z + bv.z; o.w = xv.w + bv.w;
        ((float4*)(y + r * DIM))[j4] = o;
    }

    const int wv = __builtin_amdgcn_readfirstlane(tid >> 5);  // wave id, SGPR
    const uint32_t ldsTblBase = __builtin_amdgcn_groupstaticsize();  // dyn LDS byte base
    const uint64_t gbase = (uint64_t)(uintptr_t)Yeff;

    for (int layer = 0; layer < NUM_LAYERS; ++layer) {
        __syncthreads();  // previous layer's table reads complete

#if __has_builtin(__builtin_amdgcn_tensor_load_to_lds) && \
    __has_builtin(__builtin_amdgcn_s_wait_tensorcnt)
        {
            // Each wave DMAs a 32 KB slice (8192 f32) of this layer's table.
            const uint32_t lds = ldsTblBase + (uint32_t)wv * 32768u;
            const uint64_t ga =
                gbase + (uint64_t)layer * (TBL_FLOATS * 4) + (uint64_t)wv * 32768u;
            u32x4 g0;
            g0.x = 1u;                                       // count=1, load descriptor
            g0.y = lds;                                      // lds_addr (bytes)
            g0.z = (uint32_t)ga;                             // global_addr[31:0]
            g0.w = (uint32_t)((ga >> 32) & 0x1FFFFFFu)       // global_addr[56:32]
                   | (2u << 30);                             // type=2 ("image")
            i32x8 g1;
            g1.s0 = (2 << 16);              // data_size = 4 bytes
            g1.s1 = (int)(8192u << 16);     // tensor_dim0[15:0] = 8192
            g1.s2 = (int)(1u << 16);        // tensor_dim0 hi=0 | tensor_dim1 lo=1
            g1.s3 = (int)(8192u << 16);     // tensor_dim1 hi=0 | tile_dim0 = 8192
            g1.s4 = 1;                      // tile_dim1 = 1, tile_dim2 = 0
            g1.s5 = 8192;                   // tensor_dim0_stride[31:0]
            g1.s6 = 0;                      // stride0 hi | stride1 lo
            g1.s7 = 0;                      // stride1 hi
            const i32x4 gz4 = {0, 0, 0, 0};          // groups 2/3 unused (2D tensor)
            const i32x8 gz8 = {0, 0, 0, 0, 0, 0, 0, 0};
            __builtin_amdgcn_tensor_load_to_lds(g0, g1, gz4, gz4, gz8, 0);
            __builtin_amdgcn_s_wait_tensorcnt(0);
        }
#else
        // Fallback: cooperative vector copy global -> LDS.
        {
            const float* src = Yeff + (size_t)layer * TBL_FLOATS;
            for (int v = tid; v < TBL_FLOATS / 4; v += 256)
                ((float4*)tbl)[v] = ((const float4*)src)[v];
        }
#endif
        __syncthreads();  // table visible to all waves

        // Compute: permutations folded into pair indexing.
        const int half = 1 << layer;
        for (int t = tid; t < ROWS_PER_BLOCK * PP; t += 256) {
            const int r = t >> 9;           // row in tile (PP == 512)
            const int p = t & (PP - 1);     // global pair index
            const int b = p >> layer;
            const int c = p & (half - 1);
            const int e0 = (b << (layer + 1)) + c;
            const int e1 = e0 + half;
            float* yr = y + r * DIM;
            const float x0 = yr[e0];
            const float x1 = yr[e1];
            const float u = fminf(fmaxf(x0, 0.0f), 1.0f) * 7.0f;
            const float v = fminf(fmaxf(x1, 0.0f), 1.0f) * 7.0f;
            int i0 = (int)u; i0 = i0 > 6 ? 6 : i0;
            int j0 = (int)v; j0 = j0 > 6 ? 6 : j0;
            const float du = u - (float)i0, dv = v - (float)j0;
            const float w00 = (1.0f - du) * (1.0f - dv);
            const float w10 = du * (1.0f - dv);
            const float w01 = (1.0f - du) * dv;
            const float w11 = du * dv;
            const float* tA = tbl + p * 128 + i0 * 8 + j0;  // channel 0 grid
            const float* tB = tA + 64;                      // channel 1 grid
            const float z0 = w00 * tA[0] + w10 * tA[8] + w01 * tA[1] + w11 * tA[9];
            const float z1 = w00 * tB[0] + w10 * tB[8] + w01 * tB[1] + w11 * tB[9];
            yr[e0] = x0 + z0;
            yr[e1] = x1 + z1;
        }
    }
    __syncthreads();

    // Store result rows.
    for (int v = tid; v < ROWS_PER_BLOCK * (DIM / 4); v += 256) {
        const int r = v >> 8, j4 = v & 255;
        ((float4*)(out + (row0 + r) * DIM))[j4] = ((const float4*)(y + r * DIM))[j4];
    }
}

// ---------------------------------------------------------------------------
// Launch
// ---------------------------------------------------------------------------
extern "C" void kernel_launch(void* const* d_in, const int* in_sizes, int n_in,
                              void* d_out, int out_size, void* d_ws, size_t ws_size,
                              hipStream_t stream) {
    const float* x = (const float*)d_in[0];
    const float* bias = (const float*)d_in[1];
    const float* Ys = (const float*)d_in[2];
    float* out = (float*)d_out;
    float* Yeff = (float*)d_ws;  // NUM_LAYERS*PP*2*64 floats = 2.62 MB scratch

    // Kernel 1: projection (2560 tiles of 4 grids, 1 wave each).
    const int nGrids = NUM_LAYERS * PP * 2;  // 10240
    project_wmma_kernel<<<nGrids / 4, 32, 0, stream>>>(Ys, Yeff);

    // Kernel 2: fused mixer. 320 KB dynamic LDS (table + y tile).
    const int bs = in_sizes[0] / DIM;
    const size_t smemBytes = (size_t)(TBL_FLOATS + ROWS_PER_BLOCK * DIM) * sizeof(float);
    mixer_kernel<<<bs / ROWS_PER_BLOCK, 256, smemBytes, stream>>>(x, bias, Yeff, out);
}